// GraphConvUp_22488448761963
// MI455X (gfx1250) — compile-verified
//
#include <hip/hip_runtime.h>
#include <hip/hip_bf16.h>

typedef __attribute__((ext_vector_type(2))) float v2f;
typedef __attribute__((ext_vector_type(8))) float v8f;

#define BN_EPS 1e-5f
#define BIG_D 1e8f

// ---------------------------------------------------------------------------
// KNN: per query, top-3 nearest refs (same batch), LDS-staged ref tiles.
// blockDim = 256, one thread per query. N % 256 == 0, M % 256 == 0.
// ---------------------------------------------------------------------------
__global__ void knn_kernel(const float* __restrict__ ref, const float* __restrict__ qry,
                           int* __restrict__ idx, int N, int M) {
    const int TILE = 256;
    __shared__ float4 sref[TILE];
    int q = blockIdx.x * blockDim.x + threadIdx.x;
    float4 qp = ((const float4*)qry)[q];          // (batch, x, y, z)
    float d0 = 3.4e37f, d1 = 3.4e37f, d2 = 3.4e37f;
    int   i0 = 0, i1 = 0, i2 = 0;
    for (int base = 0; base < N; base += TILE) {
        __syncthreads();
        sref[threadIdx.x] = ((const float4*)ref)[base + threadIdx.x];
        __syncthreads();
        #pragma unroll 4
        for (int j = 0; j < TILE; ++j) {
            float4 rp = sref[j];
            float dx = qp.y - rp.y, dy = qp.z - rp.z, dz = qp.w - rp.w;
            float d  = dx * dx + dy * dy + dz * dz;
            if (rp.x != qp.x) d += BIG_D;         // cross-batch penalty
            int ri = base + j;
            if (d < d0)      { d2 = d1; i2 = i1; d1 = d0; i1 = i0; d0 = d; i0 = ri; }
            else if (d < d1) { d2 = d1; i2 = i1; d1 = d;  i1 = ri; }
            else if (d < d2) { d2 = d;  i2 = ri; }
        }
    }
    if (q < M) { idx[q*3+0] = i0; idx[q*3+1] = i1; idx[q*3+2] = i2; }
}

// ---------------------------------------------------------------------------
// f32 WMMA GEMM with bias: C[Mrows x Ncols] = A[Mrows x K] * B[K x Ncols] + bias
// 256 threads = 8 waves; wave w owns 16x16 tile at (blockIdx.x*128 + w*16, blockIdx.y*16).
// Uses V_WMMA_F32_16X16X4_F32 (full fp32 precision; problem is BW-bound anyway).
// A-lane layout: lane = 16*half + l holds row M=l, K = {2*half, 2*half+1} (float2).
// B mirrored across halves; C/D: VGPR r -> row r + 8*half, col l.
// ---------------------------------------------------------------------------
__global__ void wmma_gemm_bias(const float* __restrict__ A, const float* __restrict__ B,
                               const float* __restrict__ bias, float* __restrict__ C,
                               int Mrows, int K, int Ncols) {
    int wave = threadIdx.x >> 5;
    int lane = threadIdx.x & 31;
    int half = lane >> 4;
    int l    = lane & 15;
    int tileM = (blockIdx.x * 8 + wave) * 16;
    int tileN = blockIdx.y * 16;
    if (tileM >= Mrows) return;

    int row = tileM + l;
    int col = tileN + l;

    v8f c;
    float bv = bias[col];
    #pragma unroll
    for (int r = 0; r < 8; ++r) c[r] = bv;

    const float* ap = A + (size_t)row * K + 2 * half;         // 8B aligned (K even)
    const float* bp = B + (size_t)(2 * half) * Ncols + col;

    for (int k = 0; k < K; k += 4) {
        __builtin_prefetch(ap + k + 64, 0, 0);                // global_prefetch_b8
        v2f a = *(const v2f*)(ap + k);                        // A[row][k+2h], A[row][k+2h+1]
        v2f b;
        b.x = bp[(size_t)k * Ncols];                          // B[k+2h][col]
        b.y = bp[(size_t)k * Ncols + Ncols];                  // B[k+2h+1][col]
        c = __builtin_amdgcn_wmma_f32_16x16x4_f32(
                /*neg_a=*/false, a, /*neg_b=*/false, b,
                /*c_mod=*/(short)0, c, /*reuse_a=*/false, /*reuse_b=*/false);
    }

    float* cp = C + (size_t)(tileM + 8 * half) * Ncols + col;
    #pragma unroll
    for (int r = 0; r < 8; ++r) cp[(size_t)r * Ncols] = c[r];
}

// ---------------------------------------------------------------------------
// agg[q][:] = mean over k of msg[idx[q][k]][:]  (float4-vectorized, C % 4 == 0)
// ---------------------------------------------------------------------------
__global__ void gather_mean_kernel(const float* __restrict__ msg, const int* __restrict__ idx,
                                   float* __restrict__ agg, int M, int C) {
    int vecC = C >> 2;
    int t = blockIdx.x * blockDim.x + threadIdx.x;
    if (t >= M * vecC) return;
    int q  = t / vecC;
    int c4 = t - q * vecC;
    int j0 = idx[q*3+0], j1 = idx[q*3+1], j2 = idx[q*3+2];
    const float4* m = (const float4*)msg;
    float4 a = m[(size_t)j0 * vecC + c4];
    float4 b = m[(size_t)j1 * vecC + c4];
    float4 d = m[(size_t)j2 * vecC + c4];
    float4 o;
    const float inv3 = 1.0f / 3.0f;
    o.x = (a.x + b.x + d.x) * inv3;
    o.y = (a.y + b.y + d.y) * inv3;
    o.z = (a.z + b.z + d.z) * inv3;
    o.w = (a.w + b.w + d.w) * inv3;
    ((float4*)agg)[t] = o;
}

// ---------------------------------------------------------------------------
// Per-channel mean/var (population) for both tensors.
// grid = (C, 2); blockIdx.y: 0 -> agg, 1 -> skl.
// stats layout: [mean_agg[C], var_agg[C], mean_skl[C], var_skl[C]]
// ---------------------------------------------------------------------------
__global__ void bn_stats_kernel(const float* __restrict__ agg, const float* __restrict__ skl,
                                float* __restrict__ stats, int M, int C) {
    __shared__ float ssum[256];
    __shared__ float ssq[256];
    int c = blockIdx.x;
    const float* src = blockIdx.y ? skl : agg;
    float s = 0.f, s2 = 0.f;
    for (int r = threadIdx.x; r < M; r += blockDim.x) {
        float v = src[(size_t)r * C + c];
        s += v; s2 += v * v;
    }
    ssum[threadIdx.x] = s; ssq[threadIdx.x] = s2;
    __syncthreads();
    for (int o = blockDim.x >> 1; o > 0; o >>= 1) {
        if (threadIdx.x < o) {
            ssum[threadIdx.x] += ssum[threadIdx.x + o];
            ssq[threadIdx.x]  += ssq[threadIdx.x + o];
        }
        __syncthreads();
    }
    if (threadIdx.x == 0) {
        float mean = ssum[0] / (float)M;
        float var  = ssq[0] / (float)M - mean * mean;
        stats[(blockIdx.y * 2 + 0) * C + c] = mean;
        stats[(blockIdx.y * 2 + 1) * C + c] = var;
    }
}

// ---------------------------------------------------------------------------
// out = relu( BN(agg)*g1+be1 + BN(skl)*g2+be2 )
// ---------------------------------------------------------------------------
__global__ void fuse_out_kernel(const float* __restrict__ agg, const float* __restrict__ skl,
                                const float* __restrict__ stats,
                                const float* __restrict__ g1, const float* __restrict__ be1,
                                const float* __restrict__ g2, const float* __restrict__ be2,
                                float* __restrict__ out, int M, int C) {
    int t = blockIdx.x * blockDim.x + threadIdx.x;
    if (t >= M * C) return;
    int c = t % C;
    float mc = stats[c],       vc = stats[C + c];
    float ms = stats[2*C + c], vs = stats[3*C + c];
    float a = (agg[t] - mc) * rsqrtf(vc + BN_EPS) * g1[c] + be1[c];
    float s = (skl[t] - ms) * rsqrtf(vs + BN_EPS) * g2[c] + be2[c];
    float v = a + s;
    out[t] = v > 0.f ? v : 0.f;
}

// ---------------------------------------------------------------------------
extern "C" void kernel_launch(void* const* d_in, const int* in_sizes, int n_in,
                              void* d_out, int out_size, void* d_ws, size_t ws_size,
                              hipStream_t stream) {
    const float* ref_bxyz  = (const float*)d_in[0];
    const float* ref_feat  = (const float*)d_in[1];
    const float* qry_bxyz  = (const float*)d_in[2];
    const float* skip_feat = (const float*)d_in[3];
    const float* W_conv    = (const float*)d_in[4];
    const float* b_conv    = (const float*)d_in[5];
    const float* g_conv    = (const float*)d_in[6];
    const float* be_conv   = (const float*)d_in[7];
    const float* W_skip    = (const float*)d_in[8];
    const float* b_skip    = (const float*)d_in[9];
    const float* g_skip    = (const float*)d_in[10];
    const float* be_skip   = (const float*)d_in[11];

    int N      = in_sizes[0] / 4;        // ref_bxyz is [N,4]
    int M      = in_sizes[2] / 4;        // query_bxyz is [M,4]
    int C_in   = in_sizes[1] / N;        // ref_feat [N, C_in]
    int C_skip = in_sizes[3] / M;        // query_skip_feat [M, C_skip]
    int C_up   = in_sizes[5];            // b_conv [C_up]

    // workspace carve-out (all on-chip-resident in 192MB L2)
    char* ws = (char*)d_ws;
    size_t off = 0;
    auto carve = [&](size_t bytes) -> void* {
        void* p = ws + off;
        off = (off + bytes + 255) & ~(size_t)255;
        return p;
    };
    int*   idx   = (int*)  carve((size_t)M * 3 * sizeof(int));
    float* msg   = (float*)carve((size_t)N * C_up * sizeof(float));
    float* agg   = (float*)carve((size_t)M * C_up * sizeof(float));
    float* skl   = (float*)carve((size_t)M * C_up * sizeof(float));
    float* stats = (float*)carve((size_t)4 * C_up * sizeof(float));

    // 1) KNN
    knn_kernel<<<M / 256, 256, 0, stream>>>(ref_bxyz, qry_bxyz, idx, N, M);

    // 2) msg = ref_feat @ W_conv + b_conv
    dim3 gr1((N + 127) / 128, C_up / 16);
    wmma_gemm_bias<<<gr1, 256, 0, stream>>>(ref_feat, W_conv, b_conv, msg, N, C_in, C_up);

    // 3) agg = mean(msg[idx], axis=1)
    int nv = M * (C_up / 4);
    gather_mean_kernel<<<(nv + 255) / 256, 256, 0, stream>>>(msg, idx, agg, M, C_up);

    // 4) skl = query_skip_feat @ W_skip + b_skip
    dim3 gr2((M + 127) / 128, C_up / 16);
    wmma_gemm_bias<<<gr2, 256, 0, stream>>>(skip_feat, W_skip, b_skip, skl, M, C_skip, C_up);

    // 5) BN stats for both tensors
    dim3 gs(C_up, 2);
    bn_stats_kernel<<<gs, 256, 0, stream>>>(agg, skl, stats, M, C_up);

    // 6) fused BN + BN + add + relu
    int ne = M * C_up;
    fuse_out_kernel<<<(ne + 255) / 256, 256, 0, stream>>>(agg, skl, stats,
                                                          g_conv, be_conv, g_skip, be_skip,
                                                          (float*)d_out, M, C_up);
}